// CTCLossLayer_1211180778170
// MI455X (gfx1250) — compile-verified
//
#include <hip/hip_runtime.h>

// Problem constants (match reference)
#define Bq    1024
#define Tq    256
#define Cq    100
#define Uq    48
#define Sq    97          // 2U+1
#define BLANKq 99
#define EPSq  1e-7f
#define NEGq  (-1e30f)
#define CHq   8           // rows per async chunk
#define NCHq  (Tq / CHq)  // 32 chunks

#define AS1 __attribute__((address_space(1)))
#define AS3 __attribute__((address_space(3)))

typedef int v4i __attribute__((ext_vector_type(4)));

// ---- CDNA5 async global->LDS copy (16B per active lane), ASYNCcnt tracked ----
__device__ __forceinline__ void async16(const float* g, float* l) {
  unsigned long long ga = (unsigned long long)g;
  unsigned           la = (unsigned)(unsigned long long)l;   // LDS offset = low 32 bits of generic addr
  AS1 v4i* gv = (AS1 v4i*)ga;
  AS3 v4i* lv = (AS3 v4i*)la;
#if __has_builtin(__builtin_amdgcn_global_load_async_to_lds_b128)
  __builtin_amdgcn_global_load_async_to_lds_b128(gv, lv, 0, 0);
#else
  asm volatile("global_load_async_to_lds_b128 %0, %1, off"
               :: "v"(la), "v"(ga)
               : "memory");
  (void)gv; (void)lv;
#endif
}

__device__ __forceinline__ void wait_async0() {
#if __has_builtin(__builtin_amdgcn_s_wait_asynccnt)
  __builtin_amdgcn_s_wait_asynccnt(0);
#else
  asm volatile("s_wait_asynccnt 0" ::: "memory");
#endif
}

// Issue one 8-row chunk (3200 B) of y_pred into LDS: 128 lanes * 16B + 72 lanes * 16B
__device__ __forceinline__ void issue_chunk(const float* g, float* l, int tid) {
  async16(g + 4 * tid, l + 4 * tid);                 // floats [0, 512)
  if (tid < 72)
    async16(g + 512 + 4 * tid, l + 512 + 4 * tid);   // floats [512, 800)
}

__global__ __launch_bounds__(128)
void ctc_loss_kernel(const float* __restrict__ y_pred,
                     const int*   __restrict__ y_true,
                     const int*   __restrict__ input_len,
                     const int*   __restrict__ label_len,
                     float*       __restrict__ out)
{
  __shared__ __align__(16) float lp[2][CHq * Cq];   // double-buffered 8-row chunks (2*3200 B)
  __shared__ float aA[Sq + 2];                      // alpha ping buffer, 2 leading NEG pads
  __shared__ float aB[Sq + 2];                      // alpha pong buffer
  __shared__ int   extS[Sq];

  const int b   = blockIdx.x;
  const int tid = threadIdx.x;
  const float* gbase = y_pred + (size_t)b * Tq * Cq;

  // Prefetch chunk 0 while we set up labels.
  issue_chunk(gbase, &lp[0][0], tid);

  if (tid < Sq)
    extS[tid] = (tid & 1) ? y_true[b * Uq + (tid >> 1)] : BLANKq;
  if (tid < Sq + 2) { aA[tid] = NEGq; aB[tid] = NEGq; }
  __syncthreads();   // extS + alpha init visible

  int  ext_s = BLANKq;
  bool skip  = false;
  if (tid < Sq) {
    ext_s = extS[tid];
    const int p2 = (tid >= 2) ? extS[tid - 2] : BLANKq;
    skip = (ext_s != BLANKq) && (ext_s != p2);
  }
  const int len = input_len[b];

  float* ac = aA;
  float* an = aB;

  for (int c = 0; c < NCHq; ++c) {
    wait_async0();       // each wave drains its own ASYNCcnt (chunk c complete)
    __syncthreads();     // publish chunk c LDS writes to the workgroup
    if (c + 1 < NCHq)
      issue_chunk(gbase + (size_t)(c + 1) * CHq * Cq, &lp[(c + 1) & 1][0], tid);

    const float* rows = &lp[c & 1][0];

    for (int k = 0; k < CHq; ++k) {
      const int t = c * CHq + k;
      if (tid < Sq) {
        const float lpv = __logf(rows[k * Cq + ext_s] + EPSq);   // ds_load gather + v_log_f32
        if (t == 0) {
          an[tid + 2] = (tid < 2) ? lpv : NEGq;
        } else {
          const float a0 = ac[tid + 2];
          const float a1 = ac[tid + 1];
          const float a2 = skip ? ac[tid] : NEGq;
          const float m  = fmaxf(a0, fmaxf(a1, a2));
          const float s  = __expf(a0 - m) + __expf(a1 - m) + __expf(a2 - m);
          const float nv = m + __logf(s) + lpv;
          an[tid + 2] = (t < len) ? nv : a0;
        }
      }
      __syncthreads();
      float* tmp = ac; ac = an; an = tmp;
    }
  }

  // Final reduction: ll = logsumexp(alpha[2L-1], alpha[2L]) (or alpha[0] if L==0)
  if (tid == 0) {
    const int L  = label_len[b];
    const int i2 = 2 * L;
    const int i1 = (i2 - 1 > 0) ? (i2 - 1) : 0;
    const float x1 = ac[i1 + 2];
    const float x2 = ac[i2 + 2];
    const float m  = fmaxf(x1, x2);
    const float ll = (L > 0) ? (m + __logf(__expf(x1 - m) + __expf(x2 - m)))
                             : ac[2];
    out[b] = -ll;
  }
}

extern "C" void kernel_launch(void* const* d_in, const int* in_sizes, int n_in,
                              void* d_out, int out_size, void* d_ws, size_t ws_size,
                              hipStream_t stream) {
  (void)n_in; (void)d_ws; (void)ws_size; (void)out_size;
  const float* y_pred    = (const float*)d_in[0];
  const int*   y_true    = (const int*)d_in[1];
  const int*   input_len = (const int*)d_in[2];
  const int*   label_len = (const int*)d_in[3];
  float*       out       = (float*)d_out;

  const int Bn = in_sizes[2];   // input_length has B elements
  ctc_loss_kernel<<<dim3(Bn), dim3(128), 0, stream>>>(y_pred, y_true, input_len, label_len, out);
}